// GNNLinkPredictor_28887950033462
// MI455X (gfx1250) — compile-verified
//
#include <hip/hip_runtime.h>
#include <hip/hip_bf16.h>
#include <math.h>

typedef __attribute__((ext_vector_type(2))) float v2f;
typedef __attribute__((ext_vector_type(8))) float v8f;

#define HID 64

// ---------------- degree kernels ----------------
__global__ void k_init_deg(float* deg, int n) {
  int i = blockIdx.x * blockDim.x + threadIdx.x;
  if (i < n) deg[i] = 1.0f;  // self-loop contributes 1
}

__global__ void k_count_deg(const int* __restrict__ col, float* deg, int e) {
  int i = blockIdx.x * blockDim.x + threadIdx.x;
  if (i < e) atomicAdd(&deg[col[i]], 1.0f);
}

__global__ void k_rsqrt_inplace(float* deg, int n) {
  int i = blockIdx.x * blockDim.x + threadIdx.x;
  if (i < n) deg[i] = 1.0f / sqrtf(deg[i]);  // deg >= 1 always
}

// ---------------- WMMA GEMM: C[N x 64] = A[N x K] @ B[K x 64] ----------------
// One wave per 16x16 tile; block = 4 waves covering the 64-wide N dim.
__global__ __launch_bounds__(128) void k_gemm_wmma(const float* __restrict__ A,
                                                   const float* __restrict__ Bm,
                                                   float* __restrict__ C,
                                                   int K, int Nrows) {
  const int lane = threadIdx.x & 31;
  const int wave = threadIdx.x >> 5;
  const int nl   = lane & 15;
  const int hi   = lane >> 4;            // 0: K pair {k0,k0+1}, 1: {k0+2,k0+3}
  const int m0   = blockIdx.x * 16;
  const int n0   = wave * 16;

  int m = m0 + nl;
  if (m >= Nrows) m = Nrows - 1;         // clamp (EXEC stays all-ones)
  const float* arow = A + (size_t)m * K + 2 * hi;

  v8f acc = {};
  for (int k0 = 0; k0 < K; k0 += 4) {
    v2f a = *(const v2f*)(arow + k0);                 // A[m][k0+2*hi .. +1]
    int kk = k0 + 2 * hi;
    const float* w = Bm + (size_t)kk * HID + n0 + nl;
    v2f b; b.x = w[0]; b.y = w[HID];                  // B[kk][n], B[kk+1][n]
    acc = __builtin_amdgcn_wmma_f32_16x16x4_f32(false, a, false, b,
                                                (short)0, acc, false, false);
  }
#pragma unroll
  for (int r = 0; r < 8; ++r) {
    int mr = m0 + r + 8 * hi;
    if (mr < Nrows) C[(size_t)mr * HID + n0 + nl] = acc[r];
  }
}

// ---------------- self-loop init: B[i,:] = A[i,:] * dis[i]^2 ----------------
__global__ void k_selfloop(const float* __restrict__ A, const float* __restrict__ dis,
                           float* __restrict__ B, int n) {
  int g = blockIdx.x * blockDim.x + threadIdx.x;
  if (g >= n * 16) return;
  int i = g >> 4, f0 = (g & 15) << 2;
  float s = dis[i] * dis[i];
  float4 v = *(const float4*)(A + (size_t)i * HID + f0);
  float4 o; o.x = v.x * s; o.y = v.y * s; o.z = v.z * s; o.w = v.w * s;
  *(float4*)(B + (size_t)i * HID + f0) = o;
}

// ---------------- edge aggregation: B[col] += A[row] * dis[row]*dis[col] ----
__global__ void k_edge_agg(const int* __restrict__ row, const int* __restrict__ col,
                           const float* __restrict__ dis, const float* __restrict__ A,
                           float* __restrict__ B, int e) {
  int g = blockIdx.x * blockDim.x + threadIdx.x;
  if (g >= e * 16) return;
  int ed = g >> 4, f0 = (g & 15) << 2;
  int r = row[ed], c = col[ed];
  float nrm = dis[r] * dis[c];
  float4 v = *(const float4*)(A + (size_t)r * HID + f0);
  float* d = B + (size_t)c * HID + f0;
  atomicAdd(d + 0, v.x * nrm);
  atomicAdd(d + 1, v.y * nrm);
  atomicAdd(d + 2, v.z * nrm);
  atomicAdd(d + 3, v.w * nrm);
}

// ---------------- bias (+optional ReLU) ----------------
__global__ void k_bias_act(float* __restrict__ B, const float* __restrict__ bias,
                           int n, int relu) {
  int g = blockIdx.x * blockDim.x + threadIdx.x;
  if (g >= n * HID) return;
  float v = B[g] + bias[g & (HID - 1)];
  B[g] = relu ? fmaxf(v, 0.0f) : v;
}

// ---------------- link predictor: WMMA [16 x 128] @ [128 x 64], dot, sigmoid -
__global__ __launch_bounds__(128) void k_link(const float* __restrict__ h,
                                              const int* __restrict__ pu,
                                              const int* __restrict__ pv,
                                              const float* __restrict__ lpW1,
                                              const float* __restrict__ lpb1,
                                              const float* __restrict__ lpW2,
                                              const float* __restrict__ lpb2,
                                              float* __restrict__ out, int P) {
  __shared__ float sW[2 * HID * HID];  // 128x64 fp32 = 32 KB
  for (int i = threadIdx.x; i < (2 * HID * HID) / 4; i += blockDim.x)
    ((float4*)sW)[i] = ((const float4*)lpW1)[i];
  __syncthreads();

  const int lane  = threadIdx.x & 31;
  const int wave  = threadIdx.x >> 5;
  const int nl    = lane & 15;
  const int hi    = lane >> 4;
  const int pbase = blockIdx.x * 64 + wave * 16;

  int p = pbase + nl;
  if (p >= P) p = P - 1;                 // clamp; stores guarded below
  const float* rowu = h + (size_t)pu[p] * HID;
  const float* rowv = h + (size_t)pv[p] * HID;

  v8f a0 = {}, a1 = {}, a2 = {}, a3 = {};
  for (int k0 = 0; k0 < 2 * HID; k0 += 4) {
    int kk = k0 + 2 * hi;
    const float* src = (kk < HID) ? (rowu + kk) : (rowv + kk - HID);
    v2f a = *(const v2f*)src;            // z[m][kk], z[m][kk+1]
    const float* w = sW + kk * HID + nl;
    v2f b0; b0.x = w[0];  b0.y = w[HID];
    v2f b1; b1.x = w[16]; b1.y = w[16 + HID];
    v2f b2; b2.x = w[32]; b2.y = w[32 + HID];
    v2f b3; b3.x = w[48]; b3.y = w[48 + HID];
    a0 = __builtin_amdgcn_wmma_f32_16x16x4_f32(false, a, false, b0, (short)0, a0, false, false);
    a1 = __builtin_amdgcn_wmma_f32_16x16x4_f32(false, a, false, b1, (short)0, a1, false, false);
    a2 = __builtin_amdgcn_wmma_f32_16x16x4_f32(false, a, false, b2, (short)0, a2, false, false);
    a3 = __builtin_amdgcn_wmma_f32_16x16x4_f32(false, a, false, b3, (short)0, a3, false, false);
  }

  float bb0 = lpb1[nl],      bb1 = lpb1[16 + nl];
  float bb2 = lpb1[32 + nl], bb3 = lpb1[48 + nl];
  float ww0 = lpW2[nl],      ww1 = lpW2[16 + nl];
  float ww2 = lpW2[32 + nl], ww3 = lpW2[48 + nl];
  float bias2 = lpb2[0];

#pragma unroll
  for (int r = 0; r < 8; ++r) {
    float s = fmaxf(a0[r] + bb0, 0.0f) * ww0
            + fmaxf(a1[r] + bb1, 0.0f) * ww1
            + fmaxf(a2[r] + bb2, 0.0f) * ww2
            + fmaxf(a3[r] + bb3, 0.0f) * ww3;
    // reduce across 16-lane halves (wave32): lane 0 -> row r, lane 16 -> row r+8
    s += __shfl_xor(s, 1);
    s += __shfl_xor(s, 2);
    s += __shfl_xor(s, 4);
    s += __shfl_xor(s, 8);
    if (nl == 0) {
      int pp = pbase + r + 8 * hi;
      if (pp < P) {
        float logit = s + bias2;
        out[pp] = 1.0f / (1.0f + __expf(-logit));
      }
    }
  }
}

// ---------------- host launcher ----------------
extern "C" void kernel_launch(void* const* d_in, const int* in_sizes, int n_in,
                              void* d_out, int out_size, void* d_ws, size_t ws_size,
                              hipStream_t stream) {
  const float* x    = (const float*)d_in[0];
  const int*   ei   = (const int*)d_in[1];
  const int*   ep   = (const int*)d_in[2];
  const float* W1   = (const float*)d_in[3];
  const float* b1   = (const float*)d_in[4];
  const float* W2   = (const float*)d_in[5];
  const float* b2   = (const float*)d_in[6];
  const float* lpW1 = (const float*)d_in[7];
  const float* lpb1 = (const float*)d_in[8];
  const float* lpW2 = (const float*)d_in[9];
  const float* lpb2 = (const float*)d_in[10];
  float* out = (float*)d_out;

  const int FIN = in_sizes[3] / HID;        // 128
  const int N   = in_sizes[0] / FIN;        // 100000
  const int E   = in_sizes[1] / 2;          // 1600000
  const int P   = in_sizes[2] / 2;          // 500000

  const int* row = ei;
  const int* col = ei + E;
  const int* pu  = ep;
  const int* pv  = ep + P;

  float* ws  = (float*)d_ws;
  float* dis = ws;                           // N floats
  float* A   = ws + ((N + 3) & ~3);          // N*64 floats (XW buffer)
  float* B   = A + (size_t)N * HID;          // N*64 floats (aggregated h)

  const int T = 256;
  // degrees -> deg_inv_sqrt
  k_init_deg<<<(N + T - 1) / T, T, 0, stream>>>(dis, N);
  k_count_deg<<<(E + T - 1) / T, T, 0, stream>>>(col, dis, E);
  k_rsqrt_inplace<<<(N + T - 1) / T, T, 0, stream>>>(dis, N);

  int mtiles = (N + 15) / 16;

  // ---- layer 1 ----
  k_gemm_wmma<<<mtiles, 128, 0, stream>>>(x, W1, A, FIN, N);
  k_selfloop<<<(N * 16 + T - 1) / T, T, 0, stream>>>(A, dis, B, N);
  k_edge_agg<<<(E * 16 + T - 1) / T, T, 0, stream>>>(row, col, dis, A, B, E);
  k_bias_act<<<(N * HID + T - 1) / T, T, 0, stream>>>(B, b1, N, 1);

  // ---- layer 2 ----
  k_gemm_wmma<<<mtiles, 128, 0, stream>>>(B, W2, A, HID, N);
  k_selfloop<<<(N * 16 + T - 1) / T, T, 0, stream>>>(A, dis, B, N);
  k_edge_agg<<<(E * 16 + T - 1) / T, T, 0, stream>>>(row, col, dis, A, B, E);
  k_bias_act<<<(N * HID + T - 1) / T, T, 0, stream>>>(B, b2, N, 0);

  // ---- link predictor ----
  k_link<<<(P + 63) / 64, 128, 0, stream>>>(B, pu, pv, lpW1, lpb1, lpW2, lpb2, out, P);
}